// RPN_1623497637914
// MI455X (gfx1250) — compile-verified
//
#include <hip/hip_runtime.h>
#include <hip/hip_bf16.h>

typedef __attribute__((ext_vector_type(2))) float v2f;
typedef __attribute__((ext_vector_type(8))) float v8f;

#define B_   8
#define H_   128
#define W_   128
#define A_   9
#define G_   64
#define N_   (H_ * W_ * A_)          // 147456 anchors per image
#define BLK  256
#define BLKS_PER_IMG (N_ / BLK)      // 576
#define NBLK (B_ * BLKS_PER_IMG)     // 4608 blocks total

// Precomputed standard Faster R-CNN base anchors (base_size=16, ctr=7.5,
// ratios {0.5,1,2} x scales {8,16,32}); all values are exact small integers,
// identical to the numpy round/sqrt path in the reference.
__constant__ float c_base[A_][4] = {
  { -84.f,  -40.f,  99.f,  55.f},   // r=0.5 s=8
  {-176.f,  -88.f, 191.f, 103.f},   // r=0.5 s=16
  {-360.f, -184.f, 375.f, 199.f},   // r=0.5 s=32
  { -56.f,  -56.f,  71.f,  71.f},   // r=1.0 s=8
  {-120.f, -120.f, 135.f, 135.f},   // r=1.0 s=16
  {-248.f, -248.f, 263.f, 263.f},   // r=1.0 s=32
  { -36.f,  -80.f,  51.f,  95.f},   // r=2.0 s=8
  { -80.f, -168.f,  95.f, 183.f},   // r=2.0 s=16
  {-168.f, -344.f, 183.f, 359.f}    // r=2.0 s=32
};

// Deterministic 32-lane wave sum using V_WMMA_F32_16X16X4_F32.
// A holds the per-lane values (16x4, lane-striped), B = all-ones (layout
// independent), C = 0.  D[m][*] = v[m] + v[m+16]; lanes 0..15 hold rows 0..7
// in c[0..7], lanes 16..31 hold rows 8..15, so adding the 8 accumulator
// registers and folding across the half-wave (xor 16) yields the full sum on
// every lane.  Fixed-function matrix op -> bitwise deterministic.
__device__ __forceinline__ float wave_sum32_wmma(float v) {
  v2f a; a.x = v;   a.y = 0.0f;
  v2f b; b.x = 1.0f; b.y = 1.0f;
  v8f c = {};
  c = __builtin_amdgcn_wmma_f32_16x16x4_f32(false, a, false, b,
                                            (short)0, c, false, false);
  float s = c[0] + c[1] + c[2] + c[3] + c[4] + c[5] + c[6] + c[7];
  s += __shfl_xor(s, 16, 32);
  return s;
}

__global__ __launch_bounds__(BLK)
void rpn_main_kernel(const float* __restrict__ logits,   // [B, N, 2]
                     const float* __restrict__ bbox_pred,// [B, H, W, 4A] == [B, N, 4] flat
                     const float* __restrict__ gt_boxes, // [B, G, 4]
                     const int*  __restrict__ gt_labels, // [B, G]
                     float* __restrict__ out,            // [2 + B*N + B*N*4]
                     float* __restrict__ ws)             // [3 * NBLK] partials
{
  __shared__ float s_gt[G_ * 4];
  __shared__ float s_valid[G_];
  __shared__ float s_red[3][BLK];

  const int tid = threadIdx.x;
  const int blk = blockIdx.x;
  const int b   = blk / BLKS_PER_IMG;
  const int n   = (blk % BLKS_PER_IMG) * BLK + tid;

  // Stage GT boxes + validity into LDS (256 threads cover 64*4 floats exactly).
  s_gt[tid] = gt_boxes[b * G_ * 4 + tid];
  if (tid < G_) s_valid[tid] = (gt_labels[b * G_ + tid] > 0) ? 1.0f : 0.0f;
  __syncthreads();

  const size_t idx = (size_t)b * N_ + (size_t)n;

  // Prefetch the streamed loss inputs while we grind on IoUs.
  __builtin_prefetch(&logits[idx * 2], 0, 1);
  __builtin_prefetch(&bbox_pred[idx * 4], 0, 1);

  // Anchor n: meshgrid 'ij' flattening -> cell = x*H + y.
  const int a    = n % A_;
  const int cell = n / A_;
  const int x    = cell / H_;
  const int y    = cell % H_;
  const float sx = (float)x * 16.0f;
  const float sy = (float)y * 16.0f;
  const float ax1 = c_base[a][0] + sx;
  const float ay1 = c_base[a][1] + sy;
  const float ax2 = c_base[a][2] + sx;
  const float ay2 = c_base[a][3] + sy;
  const float area1 = (ax2 - ax1) * (ay2 - ay1);

  float best = -1e30f;
  int   bg   = 0;
  float anyv = 0.0f;
#pragma unroll 8
  for (int g = 0; g < G_; ++g) {
    const float gx1 = s_gt[4 * g + 0];
    const float gy1 = s_gt[4 * g + 1];
    const float gx2 = s_gt[4 * g + 2];
    const float gy2 = s_gt[4 * g + 3];
    const float iw = fminf(ax2, gx2) - fmaxf(ax1, gx1);
    const float ih = fminf(ay2, gy2) - fmaxf(ay1, gy1);
    const float inter = fmaxf(iw, 0.0f) * fmaxf(ih, 0.0f);
    const float area2 = (gx2 - gx1) * (gy2 - gy1);
    float iou = inter / (area1 + area2 - inter + 1e-8f);
    const float v = s_valid[g];
    iou  = (v > 0.0f) ? iou : -1.0f;   // mask invalid GTs exactly like reference
    anyv += v;
    if (iou > best) { best = iou; bg = g; }  // strict '>' == argmax-first
  }

  int label = (best >= 0.7f) ? 1 : 0;

  // bbox_transform(anchor, gt[argmax])  (+1.0 width convention)
  const float mgx1 = s_gt[4 * bg + 0];
  const float mgy1 = s_gt[4 * bg + 1];
  const float mgx2 = s_gt[4 * bg + 2];
  const float mgy2 = s_gt[4 * bg + 3];
  const float bw = ax2 - ax1 + 1.0f;
  const float bh = ay2 - ay1 + 1.0f;
  const float bx = ax1 + 0.5f * bw;
  const float by = ay1 + 0.5f * bh;
  const float gw = mgx2 - mgx1 + 1.0f;
  const float gh = mgy2 - mgy1 + 1.0f;
  const float gx = mgx1 + 0.5f * gw;
  const float gy = mgy1 + 0.5f * gh;
  float t0 = (gx - bx) / bw;
  float t1 = (gy - by) / bh;
  float t2 = logf(gw / bw);
  float t3 = logf(gh / bh);
  if (anyv == 0.0f) { label = 0; t0 = t1 = t2 = t3 = 0.0f; }

  // Emit labels (as float 0/1) and targets.
  out[2 + idx] = (float)label;
  float* tg = out + 2 + (size_t)B_ * N_;
  tg[idx * 4 + 0] = t0;
  tg[idx * 4 + 1] = t1;
  tg[idx * 4 + 2] = t2;
  tg[idx * 4 + 3] = t3;

  // Softmax CE (2 classes, numerically-stable logsumexp).
  const float l0 = logits[idx * 2 + 0];
  const float l1 = logits[idx * 2 + 1];
  const float m  = fmaxf(l0, l1);
  const float lse = m + logf(expf(l0 - m) + expf(l1 - m));
  const float ce  = lse - (label ? l1 : l0);

  // Smooth-L1 on this anchor's 4 deltas.
  float sl1 = 0.0f;
  const float p0 = bbox_pred[idx * 4 + 0];
  const float p1 = bbox_pred[idx * 4 + 1];
  const float p2 = bbox_pred[idx * 4 + 2];
  const float p3 = bbox_pred[idx * 4 + 3];
  {
    float d;
    d = fabsf(p0 - t0); sl1 += (d < 1.0f) ? 0.5f * d * d : d - 0.5f;
    d = fabsf(p1 - t1); sl1 += (d < 1.0f) ? 0.5f * d * d : d - 0.5f;
    d = fabsf(p2 - t2); sl1 += (d < 1.0f) ? 0.5f * d * d : d - 0.5f;
    d = fabsf(p3 - t3); sl1 += (d < 1.0f) ? 0.5f * d * d : d - 0.5f;
  }
  const float posf = (float)label;

  // Fixed-order block tree reduction (deterministic; no float atomics).
  s_red[0][tid] = ce;
  s_red[1][tid] = sl1 * posf;
  s_red[2][tid] = posf;
  __syncthreads();
  for (int s = BLK / 2; s > 0; s >>= 1) {
    if (tid < s) {
      s_red[0][tid] += s_red[0][tid + s];
      s_red[1][tid] += s_red[1][tid + s];
      s_red[2][tid] += s_red[2][tid + s];
    }
    __syncthreads();
  }
  if (tid == 0) {
    ws[0 * NBLK + blk] = s_red[0][0];
    ws[1 * NBLK + blk] = s_red[1][0];
    ws[2 * NBLK + blk] = s_red[2][0];
  }
}

__global__ __launch_bounds__(32)
void rpn_finalize_kernel(const float* __restrict__ ws, float* __restrict__ out)
{
  const int lane = threadIdx.x;
  float pce = 0.0f, psl = 0.0f, pps = 0.0f;
  // NBLK == 144*32: no divergence, fixed per-lane accumulation order.
  for (int i = lane; i < NBLK; i += 32) {
    pce += ws[0 * NBLK + i];
    psl += ws[1 * NBLK + i];
    pps += ws[2 * NBLK + i];
  }
  // EXEC is all-ones here (full wave, no divergence) as WMMA requires.
  const float tce = wave_sum32_wmma(pce);
  const float tsl = wave_sum32_wmma(psl);
  const float tps = wave_sum32_wmma(pps);
  if (lane == 0) {
    out[0] = tce / (float)((size_t)B_ * N_);                       // all labels >= 0
    out[1] = (tps > 0.0f) ? (tsl / fmaxf(tps, 1.0f)) : 0.0f;       // npos guard
  }
}

extern "C" void kernel_launch(void* const* d_in, const int* in_sizes, int n_in,
                              void* d_out, int out_size, void* d_ws, size_t ws_size,
                              hipStream_t stream)
{
  (void)in_sizes; (void)n_in; (void)out_size; (void)ws_size;
  const float* logits    = (const float*)d_in[0];  // [B, N, 2]
  const float* bbox_pred = (const float*)d_in[1];  // [B, H, W, 36]
  const float* gt_boxes  = (const float*)d_in[2];  // [B, G, 4]
  const int*   gt_labels = (const int*)d_in[3];    // [B, G]
  float* out = (float*)d_out;
  float* ws  = (float*)d_ws;                       // needs 3*NBLK*4 = 55296 B

  rpn_main_kernel<<<NBLK, BLK, 0, stream>>>(logits, bbox_pred, gt_boxes,
                                            gt_labels, out, ws);
  rpn_finalize_kernel<<<1, 32, 0, stream>>>(ws, out);
}